// Center_59416577573137
// MI455X (gfx1250) — compile-verified
//
#include <hip/hip_runtime.h>
#include <cstdint>
#include <cstddef>

// Problem constants (from reference)
#define N_CENTER    10000
#define FEATURE_DIM 256
#define BATCH       131072
#define COEF        0.1f      // (1 - ALPHA), ALPHA = 0.9

// Tile geometry for the scatter pass
#define ROWS_PER_TILE 16                          // 16 rows x 256 f32 = 16 KB tile
#define TILE_FLOATS   (ROWS_PER_TILE * FEATURE_DIM)
#define SCATTER_BLOCKS 2048
#define SCATTER_THREADS 256                       // 8 wave32

// ---------------------------------------------------------------------------
// Pass B: stream feature tiles via async global->LDS DMA (double buffered),
// gather old center rows (L2-resident), scatter-add 0.1*(f - c) into out with
// native f32 global atomics.
// ---------------------------------------------------------------------------
__global__ void __launch_bounds__(SCATTER_THREADS)
center_scatter_kernel(const float* __restrict__ features,
                      const int*   __restrict__ labels,
                      const float* __restrict__ centers,
                      float*       __restrict__ out)
{
    __shared__ __align__(16) float lds_tile[2][TILE_FLOATS];

    const int tid   = threadIdx.x;
    const int NT    = BATCH / ROWS_PER_TILE;      // 8192 tiles
    const int step  = gridDim.x;
    int t = blockIdx.x;
    if (t >= NT) return;

    // Issue async copy of one 16 KB feature tile into lds_tile[buf].
    // 256 threads * 16 B = 4 KB per round -> 4 rounds; 4 async ops per wave.
    auto issue_tile = [&](int tile, int buf) {
        const uint64_t gbase = (uint64_t)(uintptr_t)(features + (size_t)tile * TILE_FLOATS);
        const uint32_t lbase = (uint32_t)(uintptr_t)(&lds_tile[buf][0]);
#pragma unroll
        for (int r = 0; r < 4; ++r) {
            const uint32_t off = (uint32_t)(r * SCATTER_THREADS + tid) * 16u;
            uint64_t ga = gbase + off;
            uint32_t la = lbase + off;
            asm volatile("global_load_async_to_lds_b128 %0, %1, off"
                         :: "v"(la), "v"(ga)
                         : "memory");
        }
    };

    int buf = 0;
    issue_tile(t, buf);

    for (; t < NT; t += step) {
        const int nxt = t + step;
        if (nxt < NT) {
            issue_tile(nxt, buf ^ 1);
            // This wave now has up to 8 outstanding async ops; the oldest 4
            // (current tile) complete in order -> wait until <= 4 remain.
            asm volatile("s_wait_asynccnt 4" ::: "memory");
        } else {
            asm volatile("s_wait_asynccnt 0" ::: "memory");
        }
        __syncthreads();   // all waves' DMA for this tile has landed in LDS

        const int rowbase = t * ROWS_PER_TILE;
#pragma unroll
        for (int j = 0; j < 4; ++j) {
            const int idx  = j * SCATTER_THREADS + tid;  // 0..1023 float4 slots
            const int row  = idx >> 6;                   // 16 rows
            const int col4 = idx & 63;                   // 64 float4 per row
            const int lab  = labels[rowbase + row];      // uniform per wave/j

            const float4 f = *(const float4*)&lds_tile[buf][(row << 8) + (col4 << 2)];
            const float4 c = *((const float4*)centers + ((size_t)lab << 6) + col4);

            float* o = out + (((size_t)lab << 8) + (col4 << 2));
            unsafeAtomicAdd(o + 0, COEF * (f.x - c.x));
            unsafeAtomicAdd(o + 1, COEF * (f.y - c.y));
            unsafeAtomicAdd(o + 2, COEF * (f.z - c.z));
            unsafeAtomicAdd(o + 3, COEF * (f.w - c.w));
        }
        __syncthreads();   // everyone done reading lds_tile[buf] before reuse
        buf ^= 1;
    }
}

// ---------------------------------------------------------------------------
// Launcher: Pass A is a D2D copy (out = centers), Pass B is the scatter.
// ---------------------------------------------------------------------------
extern "C" void kernel_launch(void* const* d_in, const int* in_sizes, int n_in,
                              void* d_out, int out_size, void* d_ws, size_t ws_size,
                              hipStream_t stream) {
    (void)in_sizes; (void)n_in; (void)out_size; (void)d_ws; (void)ws_size;

    const float* features = (const float*)d_in[0];
    const int*   labels   = (const int*)d_in[1];
    const float* centers  = (const float*)d_in[2];
    float*       out      = (float*)d_out;

    // Pass A: out = centers (re-initialized every call; deterministic).
    hipMemcpyAsync(out, centers,
                   (size_t)N_CENTER * FEATURE_DIM * sizeof(float),
                   hipMemcpyDeviceToDevice, stream);

    // Pass B: scatter-add.
    center_scatter_kernel<<<dim3(SCATTER_BLOCKS), dim3(SCATTER_THREADS), 0, stream>>>(
        features, labels, centers, out);
}